// DirectionalDepthwiseConv2D_37701222924349
// MI455X (gfx1250) — compile-verified
//
#include <hip/hip_runtime.h>

// Problem constants (match the reference)
#define B_N   32
#define C_N   160          // 32 central + 4*32 directional
#define H_N   128
#define W_N   128

constexpr int TILE_ROWS   = 64;             // output rows per block
constexpr int LDS_ROWS    = TILE_ROWS + 4;  // +2 halo rows top & bottom -> 68
constexpr int LDS_PITCH   = 4 + W_N + 4;    // 4-float zero pad each side -> 136 floats
constexpr int LDS_PITCH_B = LDS_PITCH * 4;  // 544 bytes (multiple of 16: B128-aligned)

// ---------------------------------------------------------------------------
// Depthwise structured conv:
//   g==0 : 3x3 centered patch          (cen_tensor[ci][3][3])
//   g==1 : horizontal 5 taps           (dir_tensor[ci][t] at (0, t-2))
//   g==2 : vertical 5 taps             (at (t-2, 0))
//   g==3 : main diagonal               (at (t-2, t-2))
//   g==4 : anti-diagonal               (at (t-2, 2-t))
// ---------------------------------------------------------------------------
__global__ __launch_bounds__(256)
void ddw_conv_kernel(const float* __restrict__ x,
                     const float* __restrict__ cen,   // [32*9]
                     const float* __restrict__ dir,   // [32*5]
                     float* __restrict__ out) {
  __shared__ float tile[LDS_ROWS * LDS_PITCH];        // 36,992 B

  const int tid = threadIdx.x;
  const int y0  = (int)blockIdx.x * TILE_ROWS;        // first output row of tile
  const int c   = (int)blockIdx.y;
  const int b   = (int)blockIdx.z;
  const int g   = c >> 5;                             // channel group
  const int ci  = c & 31;

  const int plane_elems = (b * C_N + c) * (H_N * W_N);   // < 2^27 elements
  const int plane_bytes = plane_elems * 4;               // < 2^31, fits GVS i32 offset

  const float4 z4 = make_float4(0.f, 0.f, 0.f, 0.f);

  // ---- zero the 4-float left/right pads of every LDS row (disjoint from async targets)
  if (tid < LDS_ROWS * 2) {
    const int row  = tid >> 1;
    const int side = tid & 1;
    *(float4*)&tile[row * LDS_PITCH + side * (LDS_PITCH - 4)] = z4;
  }

  // ---- stage input rows into LDS with CDNA5 async copies (ASYNCcnt path) ----
  // One wave per row: 32 lanes x 16B = 512B = one full image row.
  const int wv   = tid >> 5;
  const int lane = tid & 31;
  // Flat-aperture rule (ISA §10.2): low 32 bits of a generic LDS pointer are
  // the LDS byte offset, which is exactly what VDST of the async load needs.
  const unsigned ldsbase = (unsigned)(size_t)(const void*)tile;

  for (int j = wv; j < LDS_ROWS; j += 8) {            // wave-uniform row index
    const int y = y0 - 2 + j;                         // input row for LDS row j
    if (y >= 0 && y < H_N) {
      const unsigned ldst = ldsbase + (unsigned)(j * LDS_PITCH_B + 16 + lane * 16);
      const int goff = plane_bytes + y * (W_N * 4) + lane * 16;
      asm volatile("global_load_async_to_lds_b128 %0, %1, %2"
                   :: "v"(ldst), "v"(goff), "s"(x)
                   : "memory");
    } else {
      // out-of-image halo row: zero-fill interior (disjoint from async targets)
      *(float4*)&tile[j * LDS_PITCH + 4 + lane * 4] = z4;
    }
  }

  asm volatile("s_wait_asynccnt 0" ::: "memory");     // this wave's async copies done
  __syncthreads();                                    // cross-wave visibility

  // ---- compute: thread -> fixed column, TILE_ROWS/2 rows (stride 2) ----
  const int xc = tid & 127;                 // output column
  const int ys = tid >> 7;                  // 0 or 1
  float* outp = out + plane_elems + y0 * W_N + xc;
  const float* t0 = &tile[xc];              // interior center column = index 4

  if (g == 0) {                             // ---- 3x3 centered ----
    const float* wp = cen + ci * 9;
    const float w0 = wp[0], w1 = wp[1], w2 = wp[2];
    const float w3 = wp[3], w4 = wp[4], w5 = wp[5];
    const float w6 = wp[6], w7 = wp[7], w8 = wp[8];
    #pragma unroll 4
    for (int k = 0; k < TILE_ROWS / 2; ++k) {
      const int r = ys + 2 * k;
      const float* rm1 = t0 + (r + 1) * LDS_PITCH;
      const float* rc  = t0 + (r + 2) * LDS_PITCH;
      const float* rp1 = t0 + (r + 3) * LDS_PITCH;
      float acc;
      acc  = w0 * rm1[3] + w1 * rm1[4] + w2 * rm1[5];
      acc += w3 * rc [3] + w4 * rc [4] + w5 * rc [5];
      acc += w6 * rp1[3] + w7 * rp1[4] + w8 * rp1[5];
      outp[r * W_N] = acc;
    }
  } else {
    const float* wp = dir + ci * 5;
    const float w0 = wp[0], w1 = wp[1], w2 = wp[2], w3 = wp[3], w4 = wp[4];
    if (g == 1) {                           // ---- horizontal ----
      #pragma unroll 4
      for (int k = 0; k < TILE_ROWS / 2; ++k) {
        const int r = ys + 2 * k;
        const float* rc = t0 + (r + 2) * LDS_PITCH;
        outp[r * W_N] = w0 * rc[2] + w1 * rc[3] + w2 * rc[4]
                      + w3 * rc[5] + w4 * rc[6];
      }
    } else if (g == 2) {                    // ---- vertical ----
      #pragma unroll 4
      for (int k = 0; k < TILE_ROWS / 2; ++k) {
        const int r = ys + 2 * k;
        const float* rr = t0 + r * LDS_PITCH;
        outp[r * W_N] = w0 * rr[0 * LDS_PITCH + 4] + w1 * rr[1 * LDS_PITCH + 4]
                      + w2 * rr[2 * LDS_PITCH + 4] + w3 * rr[3 * LDS_PITCH + 4]
                      + w4 * rr[4 * LDS_PITCH + 4];
      }
    } else if (g == 3) {                    // ---- main diagonal (t-2, t-2) ----
      #pragma unroll 4
      for (int k = 0; k < TILE_ROWS / 2; ++k) {
        const int r = ys + 2 * k;
        const float* rr = t0 + r * LDS_PITCH;
        outp[r * W_N] = w0 * rr[0 * LDS_PITCH + 2] + w1 * rr[1 * LDS_PITCH + 3]
                      + w2 * rr[2 * LDS_PITCH + 4] + w3 * rr[3 * LDS_PITCH + 5]
                      + w4 * rr[4 * LDS_PITCH + 6];
      }
    } else {                                // ---- anti-diagonal (t-2, 2-t) ----
      #pragma unroll 4
      for (int k = 0; k < TILE_ROWS / 2; ++k) {
        const int r = ys + 2 * k;
        const float* rr = t0 + r * LDS_PITCH;
        outp[r * W_N] = w0 * rr[0 * LDS_PITCH + 6] + w1 * rr[1 * LDS_PITCH + 5]
                      + w2 * rr[2 * LDS_PITCH + 4] + w3 * rr[3 * LDS_PITCH + 3]
                      + w4 * rr[4 * LDS_PITCH + 2];
      }
    }
  }
}

extern "C" void kernel_launch(void* const* d_in, const int* in_sizes, int n_in,
                              void* d_out, int out_size, void* d_ws, size_t ws_size,
                              hipStream_t stream) {
  (void)in_sizes; (void)n_in; (void)out_size; (void)d_ws; (void)ws_size;
  const float* x   = (const float*)d_in[0];   // [32,160,128,128]
  const float* cen = (const float*)d_in[1];   // [32,3,3]
  const float* dir = (const float*)d_in[2];   // [32,5]
  float* out = (float*)d_out;                 // [32,160,128,128]

  dim3 grid(H_N / TILE_ROWS, C_N, B_N);       // (2, 160, 32)
  ddw_conv_kernel<<<grid, 256, 0, stream>>>(x, cen, dir, out);
}